// MeshGenLoss_60679297958331
// MI455X (gfx1250) — compile-verified
//
#include <hip/hip_runtime.h>
#include <math.h>

// ---------------------------------------------------------------------------
// MeshGen loss on MI455X (gfx1250):
//   Chamfer(pred,target) + KL(mu,logvar) + density (NN-dist std of pred)
//
// Pairwise squared-distance tiles go through the fp32 matrix pipe:
//   d2[i][j] = |p_i|^2 + |t_j|^2 + WMMA( A=p , B=-2t )   (K=3 padded to 4)
// via V_WMMA_F32_16X16X4_F32 (exact fp32 — matches the reference math).
//
// All three min-maps (Chamfer row-min, Chamfer col-min, pred NN-dist) are
// produced by the SAME row-min kernel: the col-min is just the row-min of
// the transposed pass (A=target, B=pred). Row-mins accumulate in registers
// over the wave's column chunk, so atomics happen only once per wave (8 per
// wave) instead of per tile. Diagonal masking for the self pass is a
// wave-uniform branch taken in at most one tile per wave.
// ---------------------------------------------------------------------------

typedef __attribute__((ext_vector_type(2))) float v2f;
typedef __attribute__((ext_vector_type(8))) float v8f;

#define WAVES_PER_BLOCK 8   // 256 threads = 8 wave32
#define COL_CHUNKS      4   // split of the column range per row-tile

// ---- init: fill min-buffers with +inf bit pattern -------------------------
__global__ void init_minbuf(unsigned int* __restrict__ buf, int n) {
  int i = blockIdx.x * blockDim.x + threadIdx.x;
  if (i < n) buf[i] = 0x7F800000u;  // +inf
}

// ---- row-min distance kernel ---------------------------------------------
// One wave owns a 16-row tile of P (one batch) and one column chunk of T;
// writes min_j d2(P_i, T_j) over its chunk into rowmin via atomic-min.
template <bool SELF_MASK>
__global__ __launch_bounds__(256)
void chamfer_rowmin_wmma(const float* __restrict__ P,
                         const float* __restrict__ T,
                         unsigned int* __restrict__ rowmin,
                         int Npts, int Mpts)
{
  const int rowTiles         = Npts >> 4;
  const int colTilesPerChunk = (Mpts >> 4) / COL_CHUNKS;

  const int waveId = blockIdx.x * WAVES_PER_BLOCK + (threadIdx.x >> 5);
  const int lane   = threadIdx.x & 31;
  const int half   = lane >> 4;     // 0: lanes 0-15, 1: lanes 16-31
  const int l16    = lane & 15;

  const int wavesPerBatch = rowTiles * COL_CHUNKS;
  const int b       = waveId / wavesPerBatch;
  const int rem     = waveId % wavesPerBatch;
  const int rowTile = rem / COL_CHUNKS;
  const int chunk   = rem % COL_CHUNKS;

  const int row0 = rowTile << 4;
  const float* __restrict__ Pb = P + (size_t)b * Npts * 3;
  const float* __restrict__ Tb = T + (size_t)b * Mpts * 3;

  // --- A fragment (16x4 fp32): lanes 0-15 -> (x,y), lanes 16-31 -> (z,0) ---
  const float* pp = Pb + (size_t)(row0 + l16) * 3;
  const float px = pp[0], py = pp[1], pz = pp[2];
  const float normp = px * px + py * py + pz * pz;  // |p_{row0+l16}|^2
  v2f a;
  a.x = half ? pz   : px;
  a.y = half ? 0.0f : py;

  // Row norms for the 8 D-matrix rows this lane sees (row = row0+half*8+r).
  float pn[8];
#pragma unroll
  for (int r = 0; r < 8; ++r)
    pn[r] = __shfl(normp, half * 8 + r, 32);

  float rm[8];
#pragma unroll
  for (int r = 0; r < 8; ++r) rm[r] = 3.0e38f;

  const int colTile0 = chunk * colTilesPerChunk;
#pragma unroll 2
  for (int ct = 0; ct < colTilesPerChunk; ++ct) {
    const int col0 = (colTile0 + ct) << 4;

    // --- B fragment (4x16 fp32), -2*t baked in ---
    const float* tp = Tb + (size_t)(col0 + l16) * 3;
    const float tx = tp[0], ty = tp[1], tz = tp[2];
    const float tn = tx * tx + ty * ty + tz * tz;   // |t_{col0+l16}|^2
    v2f bf;
    bf.x = half ? (-2.0f * tz) : (-2.0f * tx);
    bf.y = half ? 0.0f         : (-2.0f * ty);

    // dotm[i][j] = -2 * p_i . t_j over the 16x16 tile (fp32 matrix pipe)
    v8f c = {};
    v8f dotm = __builtin_amdgcn_wmma_f32_16x16x4_f32(
        /*neg_a=*/false, a, /*neg_b=*/false, bf,
        /*c_mod=*/(short)0, c, /*reuse_a=*/false, /*reuse_b=*/false);

    if (SELF_MASK && col0 == row0) {
      // Rare (at most one tile per wave), wave-uniform: mask the diagonal element.
#pragma unroll
      for (int r = 0; r < 8; ++r) {
        float d = fmaxf(pn[r] + tn + dotm[r], 0.0f);
        d += ((half * 8 + r) == l16) ? 1.0e6f : 0.0f;
        rm[r] = fminf(rm[r], d);
      }
    } else {
      // Hot path: no mask code at all.
#pragma unroll
      for (int r = 0; r < 8; ++r) {
        const float d = fmaxf(pn[r] + tn + dotm[r], 0.0f);
        rm[r] = fminf(rm[r], d);
      }
    }
  }

  // --- reduce row mins across the 16 lanes of each half ---
#pragma unroll
  for (int m = 1; m < 16; m <<= 1) {
#pragma unroll
    for (int r = 0; r < 8; ++r)
      rm[r] = fminf(rm[r], __shfl_xor(rm[r], m, 32));
  }
  if (l16 == 0) {
#pragma unroll
    for (int r = 0; r < 8; ++r)
      atomicMin(&rowmin[(size_t)b * Npts + row0 + half * 8 + r],
                __float_as_uint(rm[r]));
  }
}

// ---- final reductions -----------------------------------------------------
__device__ float block_reduce_sum(float v, float* red) {
  const int tid = threadIdx.x;
  red[tid] = v;
  __syncthreads();
  for (int s = 128; s > 0; s >>= 1) {
    if (tid < s) red[tid] += red[tid + s];
    __syncthreads();
  }
  const float r = red[0];
  __syncthreads();
  return r;
}

__global__ __launch_bounds__(256)
void finalize_loss(const float* __restrict__ rowmin,  // [B*N] float bits
                   const float* __restrict__ colmin,  // [B*M]
                   const float* __restrict__ nnmin,   // [B*N]
                   const float* __restrict__ mu,      // [B*L]
                   const float* __restrict__ logvar,  // [B*L]
                   float* __restrict__ out,
                   int N, int M, int L)
{
  __shared__ float red[256];
  const int tid = threadIdx.x;
  const int B = 2;

  float sRow = 0.f, sCol = 0.f;
  float sN0 = 0.f, sN1 = 0.f, sQ0 = 0.f, sQ1 = 0.f;
  float sKL = 0.f;

  for (int i = tid; i < B * N; i += 256) sRow += rowmin[i];
  for (int i = tid; i < B * M; i += 256) sCol += colmin[i];
  for (int i = tid; i < N; i += 256) {
    const float a0 = nnmin[i];     sN0 += a0; sQ0 += a0 * a0;
    const float a1 = nnmin[N + i]; sN1 += a1; sQ1 += a1 * a1;
  }
  for (int i = tid; i < B * L; i += 256) {
    const float m = mu[i], lv = logvar[i];
    sKL += 1.0f + lv - m * m - expf(lv);
  }

  sRow = block_reduce_sum(sRow, red);
  sCol = block_reduce_sum(sCol, red);
  sN0  = block_reduce_sum(sN0, red);
  sN1  = block_reduce_sum(sN1, red);
  sQ0  = block_reduce_sum(sQ0, red);
  sQ1  = block_reduce_sum(sQ1, red);
  sKL  = block_reduce_sum(sKL, red);

  if (tid == 0) {
    const float cd = sRow / ((float)N * B) + sCol / ((float)M * B);
    const float var0 = (sQ0 - sN0 * sN0 / (float)N) / (float)(N - 1);
    const float var1 = (sQ1 - sN1 * sN1 / (float)N) / (float)(N - 1);
    const float density = 0.5f * (sqrtf(fmaxf(var0, 0.f)) +
                                  sqrtf(fmaxf(var1, 0.f)));
    const float kl = -0.5f * sKL / ((float)B * (float)L);
    const float total = cd + 0.001f * kl + 0.1f * density;
    out[0] = total;
    out[1] = cd;
    out[2] = kl;
    out[3] = density;
  }
}

// ---------------------------------------------------------------------------
extern "C" void kernel_launch(void* const* d_in, const int* in_sizes, int n_in,
                              void* d_out, int out_size, void* d_ws, size_t ws_size,
                              hipStream_t stream) {
  const float* pred   = (const float*)d_in[0];  // [B,N,3]
  const float* target = (const float*)d_in[1];  // [B,M,3]
  const float* mu     = (const float*)d_in[2];  // [B,L]
  const float* logvar = (const float*)d_in[3];  // [B,L]

  const int B = 2;
  const int N = in_sizes[0] / (B * 3);
  const int M = in_sizes[1] / (B * 3);
  const int L = in_sizes[2] / B;

  unsigned int* rowmin = (unsigned int*)d_ws;       // [B*N] min_j d2(pred_i, target_j)
  unsigned int* colmin = rowmin + (size_t)B * N;    // [B*M] min_i d2(target_j, pred_i)
  unsigned int* nnmin  = colmin + (size_t)B * M;    // [B*N] pred NN dists (diag masked)
  const int totalMin = B * N + B * M + B * N;

  init_minbuf<<<(totalMin + 255) / 256, 256, 0, stream>>>(rowmin, totalMin);

  // pred -> target  (Chamfer term 1: min over M per pred point)
  {
    const int waves = B * (N / 16) * COL_CHUNKS;
    chamfer_rowmin_wmma<false><<<waves / WAVES_PER_BLOCK, 256, 0, stream>>>(
        pred, target, rowmin, N, M);
  }
  // target -> pred  (Chamfer term 2: min over N per target point)
  {
    const int waves = B * (M / 16) * COL_CHUNKS;
    chamfer_rowmin_wmma<false><<<waves / WAVES_PER_BLOCK, 256, 0, stream>>>(
        target, pred, colmin, M, N);
  }
  // pred -> pred    (density term: NN distance, diagonal masked)
  {
    const int waves = B * (N / 16) * COL_CHUNKS;
    chamfer_rowmin_wmma<true><<<waves / WAVES_PER_BLOCK, 256, 0, stream>>>(
        pred, pred, nnmin, N, N);
  }

  finalize_loss<<<1, 256, 0, stream>>>((const float*)rowmin,
                                       (const float*)colmin,
                                       (const float*)nnmin,
                                       mu, logvar, (float*)d_out, N, M, L);
}